// MambaBlock_v1_1520418422822
// MI455X (gfx1250) — compile-verified
//
#include <hip/hip_runtime.h>
#include <hip/hip_bf16.h>
#include <math.h>

// ---------------------------------------------------------------------------
// Mamba block (selective SSM) + GroupNorm + Mish for gfx1250 (MI455X, wave32)
// GEMMs run on V_WMMA_F32_16X16X4_F32 (full fp32 matrix pipe).
// Each wave computes a 16x64 output macro-tile (1 M-tile x 4 N-tiles) so one
// A fragment feeds four WMMAs per K-step.  Prefetches are unconditional
// (speculative global_prefetch_b8) to keep the inner loop branch-free.
// ---------------------------------------------------------------------------

typedef float v2f __attribute__((ext_vector_type(2)));
typedef float v8f __attribute__((ext_vector_type(8)));

#define BATCH    8
#define SEQ      512
#define DMODEL   512
#define DINNER   1024
#define DSTATE   16
#define DTRANK   32
#define NGROUPS  8
#define MROWS    (BATCH * SEQ)        // 4096

__device__ __forceinline__ v8f wmma_f32_4(v2f a, v2f b, v8f c) {
    // 8 args: (neg_a, A, neg_b, B, c_mod, C, reuse_a, reuse_b)
    return __builtin_amdgcn_wmma_f32_16x16x4_f32(
        false, a, false, b, (short)0, c, false, false);
}

__device__ __forceinline__ float softplus_f(float x) {
    return fmaxf(x, 0.f) + log1pf(__expf(-fabsf(x)));
}

// ---------------------------------------------------------------------------
// Kernel 1: in_proj  xz[m, n] = sum_k x[b, k, l] * W[n, k]
//   M = 4096 (m = b*512 + l), N = 2048, K = 512.  x is [B, D(=K), L].
// Wave = 16x64 macro-tile: A fragment (strided gather from x) reused by 4 WMMAs.
// ---------------------------------------------------------------------------
__global__ void k_inproj(const float* __restrict__ x,
                         const float* __restrict__ w,
                         float* __restrict__ xz) {
    const int wid   = blockIdx.x * 4 + (threadIdx.x >> 5);   // 8192 waves
    const int ntq   = wid & 31;           // N quad-tile: 2048/64 = 32
    const int tileM = wid >> 5;           // 4096/16 = 256
    const int lane  = threadIdx.x & 31;
    const int half  = lane >> 4;
    const int r     = lane & 15;

    const int m  = tileM * 16 + r;        // A-fragment row
    const int bb = m >> 9;
    const int ll = m & 511;

    const float* xb = x + (size_t)bb * DMODEL * SEQ + ll;    // + k*SEQ walks K
    const float* wr0 = w + (size_t)((ntq * 4 + 0) * 16 + r) * DMODEL;
    const float* wr1 = w + (size_t)((ntq * 4 + 1) * 16 + r) * DMODEL;
    const float* wr2 = w + (size_t)((ntq * 4 + 2) * 16 + r) * DMODEL;
    const float* wr3 = w + (size_t)((ntq * 4 + 3) * 16 + r) * DMODEL;

    v8f acc0 = {}, acc1 = {}, acc2 = {}, acc3 = {};
    #pragma unroll 2
    for (int k0 = 0; k0 < DMODEL; k0 += 4) {
        const int ka = k0 + half * 2;
        v2f a;
        a.x = xb[(size_t)ka * SEQ];
        a.y = xb[(size_t)(ka + 1) * SEQ];
        v2f b0 = *(const v2f*)(wr0 + ka);
        v2f b1 = *(const v2f*)(wr1 + ka);
        v2f b2 = *(const v2f*)(wr2 + ka);
        v2f b3 = *(const v2f*)(wr3 + ka);
        __builtin_prefetch(wr0 + k0 + 32, 0, 1);   // speculative, branch-free
        acc0 = wmma_f32_4(a, b0, acc0);
        acc1 = wmma_f32_4(a, b1, acc1);
        acc2 = wmma_f32_4(a, b2, acc2);
        acc3 = wmma_f32_4(a, b3, acc3);
    }
    #pragma unroll
    for (int j = 0; j < 8; ++j) {
        const size_t row = (size_t)(tileM * 16 + half * 8 + j) * (2 * DINNER);
        xz[row + ntq * 64 + 0  + r] = acc0[j];
        xz[row + ntq * 64 + 16 + r] = acc1[j];
        xz[row + ntq * 64 + 32 + r] = acc2[j];
        xz[row + ntq * 64 + 48 + r] = acc3[j];
    }
}

// ---------------------------------------------------------------------------
// Kernel 2: causal depthwise conv (k=4) + bias + SiLU on u = xz[:, :1024]
// ---------------------------------------------------------------------------
__global__ void k_conv(const float* __restrict__ xz,
                       const float* __restrict__ cw,
                       const float* __restrict__ cb,
                       float* __restrict__ ucv) {
    const int idx = blockIdx.x * blockDim.x + threadIdx.x;
    if (idx >= MROWS * DINNER) return;
    const int c = idx & (DINNER - 1);
    const int m = idx >> 10;              // b*512 + l
    const int l = m & 511;

    float s = cb[c];
    #pragma unroll
    for (int t = 0; t < 4; ++t) {
        const int lt = l - 3 + t;
        if (lt >= 0)
            s += xz[(size_t)(m - 3 + t) * (2 * DINNER) + c] * cw[c * 4 + t];
    }
    const float sig = 1.f / (1.f + __expf(-s));
    ucv[(size_t)m * DINNER + c] = s * sig;
}

// ---------------------------------------------------------------------------
// Kernel 3: x_proj  dbc[m, n] = sum_k ucv[m, k] * Wx[n, k]
//   M = 4096, N = 64, K = 1024.  One wave covers the whole N (4 tiles).
// ---------------------------------------------------------------------------
__global__ void k_xproj(const float* __restrict__ ucv,
                        const float* __restrict__ w,
                        float* __restrict__ dbc) {
    const int tileM = blockIdx.x * 4 + (threadIdx.x >> 5);   // 256 waves
    const int lane  = threadIdx.x & 31;
    const int half  = lane >> 4;
    const int r     = lane & 15;

    const float* arow = ucv + (size_t)(tileM * 16 + r) * DINNER;
    const float* wr0 = w + (size_t)(0  + r) * DINNER;
    const float* wr1 = w + (size_t)(16 + r) * DINNER;
    const float* wr2 = w + (size_t)(32 + r) * DINNER;
    const float* wr3 = w + (size_t)(48 + r) * DINNER;

    v8f acc0 = {}, acc1 = {}, acc2 = {}, acc3 = {};
    #pragma unroll 2
    for (int k0 = 0; k0 < DINNER; k0 += 4) {
        const int ka = k0 + half * 2;
        v2f a  = *(const v2f*)(arow + ka);
        v2f b0 = *(const v2f*)(wr0 + ka);
        v2f b1 = *(const v2f*)(wr1 + ka);
        v2f b2 = *(const v2f*)(wr2 + ka);
        v2f b3 = *(const v2f*)(wr3 + ka);
        __builtin_prefetch(arow + k0 + 32, 0, 1);
        acc0 = wmma_f32_4(a, b0, acc0);
        acc1 = wmma_f32_4(a, b1, acc1);
        acc2 = wmma_f32_4(a, b2, acc2);
        acc3 = wmma_f32_4(a, b3, acc3);
    }
    #pragma unroll
    for (int j = 0; j < 8; ++j) {
        const size_t row = (size_t)(tileM * 16 + half * 8 + j) * 64;
        dbc[row + 0  + r] = acc0[j];
        dbc[row + 16 + r] = acc1[j];
        dbc[row + 32 + r] = acc2[j];
        dbc[row + 48 + r] = acc3[j];
    }
}

// ---------------------------------------------------------------------------
// Kernel 4: dt_proj + softplus
//   delta[m, n] = softplus( sum_k dbc[m, k] * Wd[n, k] + bias[n] )
//   M = 4096, N = 1024, K = 32 (dt = dbc[:, :32], row stride 64).
// ---------------------------------------------------------------------------
__global__ void k_dtproj(const float* __restrict__ dbc,
                         const float* __restrict__ w,
                         const float* __restrict__ bias,
                         float* __restrict__ delta) {
    const int wid   = blockIdx.x * 4 + (threadIdx.x >> 5);   // 4096 waves
    const int ntq   = wid & 15;           // 1024/64 = 16 quads
    const int tileM = wid >> 4;
    const int lane  = threadIdx.x & 31;
    const int half  = lane >> 4;
    const int r     = lane & 15;

    const float* arow = dbc + (size_t)(tileM * 16 + r) * 64;
    const float* wr0 = w + (size_t)((ntq * 4 + 0) * 16 + r) * DTRANK;
    const float* wr1 = w + (size_t)((ntq * 4 + 1) * 16 + r) * DTRANK;
    const float* wr2 = w + (size_t)((ntq * 4 + 2) * 16 + r) * DTRANK;
    const float* wr3 = w + (size_t)((ntq * 4 + 3) * 16 + r) * DTRANK;

    v8f acc0 = {}, acc1 = {}, acc2 = {}, acc3 = {};
    #pragma unroll
    for (int k0 = 0; k0 < DTRANK; k0 += 4) {
        const int ka = k0 + half * 2;
        v2f a  = *(const v2f*)(arow + ka);
        v2f b0 = *(const v2f*)(wr0 + ka);
        v2f b1 = *(const v2f*)(wr1 + ka);
        v2f b2 = *(const v2f*)(wr2 + ka);
        v2f b3 = *(const v2f*)(wr3 + ka);
        acc0 = wmma_f32_4(a, b0, acc0);
        acc1 = wmma_f32_4(a, b1, acc1);
        acc2 = wmma_f32_4(a, b2, acc2);
        acc3 = wmma_f32_4(a, b3, acc3);
    }
    const int nb = ntq * 64 + r;
    const float bn0 = bias[nb + 0];
    const float bn1 = bias[nb + 16];
    const float bn2 = bias[nb + 32];
    const float bn3 = bias[nb + 48];
    #pragma unroll
    for (int j = 0; j < 8; ++j) {
        const size_t row = (size_t)(tileM * 16 + half * 8 + j) * DINNER;
        delta[row + nb + 0 ] = softplus_f(acc0[j] + bn0);
        delta[row + nb + 16] = softplus_f(acc1[j] + bn1);
        delta[row + nb + 32] = softplus_f(acc2[j] + bn2);
        delta[row + nb + 48] = softplus_f(acc3[j] + bn3);
    }
}

// ---------------------------------------------------------------------------
// Kernel 5: selective scan + skip + SiLU gate.
// Grid (8 channel-groups x B batches), 128 threads: thread = one channel.
// h[16] state in registers; B_t / C_t broadcast via LDS each step.
// ---------------------------------------------------------------------------
__global__ void k_scan(const float* __restrict__ delta,
                       const float* __restrict__ ucv,
                       const float* __restrict__ xz,
                       const float* __restrict__ dbc,
                       const float* __restrict__ A_log,
                       const float* __restrict__ D_param,
                       float* __restrict__ ybuf) {
    const int bb = blockIdx.x >> 3;                  // batch
    const int c  = (blockIdx.x & 7) * 128 + threadIdx.x;
    const int tid = threadIdx.x;
    const int mbase = bb * SEQ;

    __shared__ float sB[DSTATE];
    __shared__ float sC[DSTATE];

    float Ac[DSTATE];
    float h[DSTATE];
    #pragma unroll
    for (int n = 0; n < DSTATE; ++n) {
        Ac[n] = -__expf(A_log[(size_t)c * DSTATE + n]);
        h[n] = 0.f;
    }
    const float Dp = D_param[c];

    for (int l = 0; l < SEQ; ++l) {
        const int m = mbase + l;
        __syncthreads();
        if (tid < 16)       sB[tid]      = dbc[(size_t)m * 64 + 32 + tid];
        else if (tid < 32)  sC[tid - 16] = dbc[(size_t)m * 64 + 48 + (tid - 16)];
        __syncthreads();

        const float dlt = delta[(size_t)m * DINNER + c];
        const float ut  = ucv[(size_t)m * DINNER + c];
        const float zt  = xz[(size_t)m * (2 * DINNER) + DINNER + c];
        const float du  = dlt * ut;

        float y = 0.f;
        #pragma unroll
        for (int n = 0; n < DSTATE; ++n) {
            const float dA = __expf(dlt * Ac[n]);
            h[n] = dA * h[n] + du * sB[n];
            y += h[n] * sC[n];
        }
        const float sig = 1.f / (1.f + __expf(-zt));
        ybuf[(size_t)m * DINNER + c] = (y + ut * Dp) * (zt * sig);
    }
}

// ---------------------------------------------------------------------------
// Kernel 6: out_proj   otmp[b, n, l] = sum_k ybuf[m, k] * Wo[n, k]
//   M = 4096, N = 512, K = 1024.  Stored directly in [B, C, L] layout.
// ---------------------------------------------------------------------------
__global__ void k_outproj(const float* __restrict__ ybuf,
                          const float* __restrict__ w,
                          float* __restrict__ otmp) {
    const int wid   = blockIdx.x * 4 + (threadIdx.x >> 5);   // 2048 waves
    const int ntq   = wid & 7;            // 512/64 = 8 quads
    const int tileM = wid >> 3;
    const int lane  = threadIdx.x & 31;
    const int half  = lane >> 4;
    const int r     = lane & 15;

    const float* arow = ybuf + (size_t)(tileM * 16 + r) * DINNER;
    const float* wr0 = w + (size_t)((ntq * 4 + 0) * 16 + r) * DINNER;
    const float* wr1 = w + (size_t)((ntq * 4 + 1) * 16 + r) * DINNER;
    const float* wr2 = w + (size_t)((ntq * 4 + 2) * 16 + r) * DINNER;
    const float* wr3 = w + (size_t)((ntq * 4 + 3) * 16 + r) * DINNER;

    v8f acc0 = {}, acc1 = {}, acc2 = {}, acc3 = {};
    #pragma unroll 2
    for (int k0 = 0; k0 < DINNER; k0 += 4) {
        const int ka = k0 + half * 2;
        v2f a  = *(const v2f*)(arow + ka);
        v2f b0 = *(const v2f*)(wr0 + ka);
        v2f b1 = *(const v2f*)(wr1 + ka);
        v2f b2 = *(const v2f*)(wr2 + ka);
        v2f b3 = *(const v2f*)(wr3 + ka);
        __builtin_prefetch(arow + k0 + 32, 0, 1);
        acc0 = wmma_f32_4(a, b0, acc0);
        acc1 = wmma_f32_4(a, b1, acc1);
        acc2 = wmma_f32_4(a, b2, acc2);
        acc3 = wmma_f32_4(a, b3, acc3);
    }
    #pragma unroll
    for (int j = 0; j < 8; ++j) {
        const int m = tileM * 16 + half * 8 + j;    // b*512 + l
        const int b = m >> 9;
        const int l = m & 511;
        const size_t base = ((size_t)b * DMODEL + ntq * 64 + r) * SEQ + l;
        otmp[base + 0  * SEQ] = acc0[j];
        otmp[base + 16 * SEQ] = acc1[j];
        otmp[base + 32 * SEQ] = acc2[j];
        otmp[base + 48 * SEQ] = acc3[j];
    }
}

// ---------------------------------------------------------------------------
// Kernel 7: GroupNorm statistics.  One block per (b, group): contiguous
// 64 channels * 512 L = 32768 floats of otmp.
// ---------------------------------------------------------------------------
__global__ void k_gnstats(const float* __restrict__ otmp,
                          float* __restrict__ stats) {
    const int bid = blockIdx.x;           // 0 .. 63
    const int tid = threadIdx.x;
    const float* base = otmp + (size_t)bid * 32768;

    __shared__ float ssum[256];
    __shared__ float ssq[256];
    float s = 0.f, q = 0.f;
    for (int i = tid; i < 32768; i += 256) {
        const float v = base[i];
        s += v;
        q += v * v;
    }
    ssum[tid] = s; ssq[tid] = q;
    __syncthreads();
    for (int off = 128; off > 0; off >>= 1) {
        if (tid < off) { ssum[tid] += ssum[tid + off]; ssq[tid] += ssq[tid + off]; }
        __syncthreads();
    }
    if (tid == 0) {
        const float mu  = ssum[0] * (1.f / 32768.f);
        const float var = ssq[0] * (1.f / 32768.f) - mu * mu;
        stats[2 * bid]     = mu;
        stats[2 * bid + 1] = rsqrtf(var + 1e-5f);
    }
}

// ---------------------------------------------------------------------------
// Kernel 8: normalize + affine + Mish -> d_out  ([B, D, L])
// ---------------------------------------------------------------------------
__global__ void k_gnmish(const float* __restrict__ otmp,
                         const float* __restrict__ stats,
                         const float* __restrict__ gw,
                         const float* __restrict__ gb,
                         float* __restrict__ out) {
    const int idx = blockIdx.x * blockDim.x + threadIdx.x;
    if (idx >= BATCH * DMODEL * SEQ) return;
    const int b   = idx >> 18;            // / (512*512)
    const int rem = idx & (DMODEL * SEQ - 1);
    const int c   = rem >> 9;
    const int g   = c >> 6;               // / 64
    const int blk = b * NGROUPS + g;

    const float mu = stats[2 * blk];
    const float rs = stats[2 * blk + 1];
    float v = (otmp[idx] - mu) * rs * gw[c] + gb[c];
    out[idx] = v * tanhf(softplus_f(v));
}

// ---------------------------------------------------------------------------
// Host launcher
// ---------------------------------------------------------------------------
extern "C" void kernel_launch(void* const* d_in, const int* in_sizes, int n_in,
                              void* d_out, int out_size, void* d_ws, size_t ws_size,
                              hipStream_t stream) {
    const float* x         = (const float*)d_in[0];
    const float* in_proj_w = (const float*)d_in[1];
    const float* conv_w    = (const float*)d_in[2];
    const float* conv_b    = (const float*)d_in[3];
    const float* x_proj_w  = (const float*)d_in[4];
    const float* dt_proj_w = (const float*)d_in[5];
    const float* dt_proj_b = (const float*)d_in[6];
    const float* A_log     = (const float*)d_in[7];
    const float* D_param   = (const float*)d_in[8];
    const float* out_proj_w= (const float*)d_in[9];
    const float* gn_w      = (const float*)d_in[10];
    const float* gn_b      = (const float*)d_in[11];
    float* out = (float*)d_out;

    // Workspace layout (floats).  otmp reuses the xz region: the full xz
    // buffer is dead after the scan kernel completes (stream-ordered).
    float* ws    = (float*)d_ws;
    float* xz    = ws;                                 //  8,388,608  (4096 x 2048)
    float* ucv   = xz    + (size_t)MROWS * 2 * DINNER; //  4,194,304  (4096 x 1024)
    float* dbc   = ucv   + (size_t)MROWS * DINNER;     //    262,144  (4096 x 64)
    float* delta = dbc   + (size_t)MROWS * 64;         //  4,194,304
    float* ybuf  = delta + (size_t)MROWS * DINNER;     //  4,194,304
    float* stats = ybuf  + (size_t)MROWS * DINNER;     //        128
    float* otmp  = xz;                                 //  2,097,152  (reuse)

    // 1) in_proj GEMM: 256 M-tiles x 32 N-quads = 8192 waves, 4 waves/block
    k_inproj<<<8192 / 4, 128, 0, stream>>>(x, in_proj_w, xz);

    // 2) depthwise causal conv + SiLU
    k_conv<<<(MROWS * DINNER + 255) / 256, 256, 0, stream>>>(xz, conv_w, conv_b, ucv);

    // 3) x_proj GEMM: 256 waves (each covers full N=64)
    k_xproj<<<256 / 4, 128, 0, stream>>>(ucv, x_proj_w, dbc);

    // 4) dt_proj GEMM + softplus: 256 x 16 quads = 4096 waves
    k_dtproj<<<4096 / 4, 128, 0, stream>>>(dbc, dt_proj_w, dt_proj_b, delta);

    // 5) selective scan: 8 batches x 8 channel-groups, 128 threads each
    k_scan<<<BATCH * 8, 128, 0, stream>>>(delta, ucv, xz, dbc, A_log, D_param, ybuf);

    // 6) out_proj GEMM -> [B, C, L]: 256 x 8 quads = 2048 waves
    k_outproj<<<2048 / 4, 128, 0, stream>>>(ybuf, out_proj_w, otmp);

    // 7) GroupNorm stats: one block per (batch, group)
    k_gnstats<<<BATCH * NGROUPS, 256, 0, stream>>>(otmp, stats);

    // 8) normalize + affine + Mish
    k_gnmish<<<(BATCH * DMODEL * SEQ + 255) / 256, 256, 0, stream>>>(
        otmp, stats, gn_w, gn_b, out);
}